// MorphologicalErosion_62861141344689
// MI455X (gfx1250) — compile-verified
//
#include <hip/hip_runtime.h>
#include <hip/hip_bf16.h>
#include <cstdint>

// Morphological erosion: out[b,i,j,f] = min_{di,dj,c} ( x[b,i+di,j+dj,c] - W[di,dj,c,f] )
// x: (16,128,128,16) f32 NHWC, W: (3,3,16,32) f32, out: (16,126,126,32) f32.
// Tropical (min,+) conv => WMMA not applicable; VALU-bound; CDNA5 path = TDM LDS staging + v_min3.

#define BATCH 16
#define HIN   128
#define WIN   128
#define CIN   16
#define FOUT  32
#define HOUT  126
#define WOUT  126
#define TILE  14          // output tile is TILE x TILE, input tile 16x16
#define NTIL  9           // 9*14 = 126

#if defined(__HIP_DEVICE_COMPILE__) && \
    __has_builtin(__builtin_amdgcn_tensor_load_to_lds) && \
    __has_builtin(__builtin_amdgcn_s_wait_tensorcnt)
#define USE_TDM 1
#else
#define USE_TDM 0
#endif

#if USE_TDM
typedef unsigned int u32x4 __attribute__((ext_vector_type(4)));
typedef int          i32x4 __attribute__((ext_vector_type(4)));
typedef int          i32x8 __attribute__((ext_vector_type(8)));

typedef const __attribute__((address_space(3))) char* lds_cptr_t;

__device__ __forceinline__ unsigned lds_byte_offset(const void* p) {
    // generic -> LDS addrspace cast; AS(3) pointers are 32-bit LDS byte offsets
    return (unsigned)(unsigned long long)(lds_cptr_t)p;
}

// 2-D TDM tile load, f32 elements, built per CDNA5 D# bitfields (ISA ch. 10.7/8).
__device__ __forceinline__ void tdm_load_tile_2d(
    unsigned lds_addr, const void* gptr,
    unsigned tensor_d0, unsigned tensor_d1,
    unsigned tile_d0, unsigned tile_d1,
    unsigned long long stride0_elems)
{
    unsigned long long ga = (unsigned long long)gptr;
    u32x4 g0;
    g0.x = 1u;                                               // count = 1 valid descriptor
    g0.y = lds_addr;                                         // lds_addr [63:32]
    g0.z = (unsigned)(ga & 0xFFFFFFFFull);                   // global_addr lo
    g0.w = (unsigned)((ga >> 32) & 0x1FFFFFFull)             // global_addr [56:32]
         | (2u << 30);                                       // type = 2 ("image")
    i32x8 g1;
    g1[0] = (int)(2u << 16);                                 // data_size = 2 -> 4 bytes
    g1[1] = (int)((tensor_d0 & 0xFFFFu) << 16);              // tensor_dim0 [63:48]
    g1[2] = (int)(((tensor_d0 >> 16) & 0xFFFFu)              // tensor_dim0 [79:64]
                | ((tensor_d1 & 0xFFFFu) << 16));            // tensor_dim1 [95:80]
    g1[3] = (int)(((tensor_d1 >> 16) & 0xFFFFu)              // tensor_dim1 [111:96]
                | ((tile_d0 & 0xFFFFu) << 16));              // tile_dim0  [127:112]
    g1[4] = (int)(tile_d1 & 0xFFFFu);                        // tile_dim1 [143:128], tile_dim2 = 0
    g1[5] = (int)(stride0_elems & 0xFFFFFFFFull);            // tensor_dim0_stride lo32
    g1[6] = (int)((stride0_elems >> 32) & 0xFFFFull);        // stride0 hi16, stride1 = 0
    g1[7] = 0;
    i32x4 gz4 = {0, 0, 0, 0};                                // groups 2/3 unused (<=2D)
    i32x8 gz8 = {0, 0, 0, 0, 0, 0, 0, 0};
    // 6-arg form (this toolchain): (g0, g1, g2, g3, g_extra, cpol)
    __builtin_amdgcn_tensor_load_to_lds(g0, g1, gz4, gz4, gz8, 0);
}
#endif

__device__ __forceinline__ float min3f(float a, float b, float c) {
    return __builtin_fminf(__builtin_fminf(a, b), c);        // -> v_min3_num_f32
}

__global__ __launch_bounds__(256)
void MorphologicalErosion_62861141344689_kernel(const float* __restrict__ x,
                                                const float* __restrict__ w,
                                                float* __restrict__ out)
{
    __shared__ __align__(16) float xs[16 * 16 * CIN];        // 16 KB: x tile [ii][jj][c]
    __shared__ __align__(16) float wl[3 * 3 * CIN * FOUT];   // 18 KB: W [kpos][c][f]

    const int tid = threadIdx.x;
    const int b   = blockIdx.z;
    const int i0  = blockIdx.y * TILE;
    const int j0  = blockIdx.x * TILE;

    const size_t xbase = (((size_t)b * HIN + i0) * WIN + j0) * CIN;   // in floats

#if USE_TDM
    if (tid < 32) {   // wave 0 issues both DMA tile loads
        // W: contiguous 4608 f32 as a 1-row tile
        tdm_load_tile_2d(lds_byte_offset(wl), w,
                         3 * 3 * CIN * FOUT, 1,
                         3 * 3 * CIN * FOUT, 1,
                         3 * 3 * CIN * FOUT);
        // x tile: 16 rows of 256 contiguous f32 (16 j * 16 c), row stride = 128*16 f32
        tdm_load_tile_2d(lds_byte_offset(xs), x + xbase,
                         WIN * CIN, HIN,
                         16 * CIN, 16,
                         WIN * CIN);
        __builtin_amdgcn_s_wait_tensorcnt(0);
    }
#else
    for (int t = tid; t < 3 * 3 * CIN * FOUT; t += 256) wl[t] = w[t];
    {
        const float4* xg  = reinterpret_cast<const float4*>(x);
        float4*       xs4 = reinterpret_cast<float4*>(xs);
        for (int q = tid; q < 16 * 16 * CIN / 4; q += 256) {
            const int ii = q >> 6;          // row in tile (64 float4 per row)
            const int r  = q & 63;
            xs4[q] = xg[(xbase >> 2) + (size_t)ii * (WIN * CIN / 4) + r];
        }
    }
#endif
    __syncthreads();

    const int fg = tid & 7;          // f-group: f = fg*4 .. fg*4+3
    const int sp = tid >> 3;         // spatial lane 0..31

    const float4* XS4 = reinterpret_cast<const float4*>(xs);
    const float4* WL4 = reinterpret_cast<const float4*>(wl);

    float4 acc[7];
    int    xb[7];                    // float4 base index of (pi,pj) in xs
    const float inf = __builtin_inff();
    #pragma unroll
    for (int k = 0; k < 7; ++k) {
        int p  = sp + 32 * k;
        int pc = (p < TILE * TILE) ? p : (TILE * TILE - 1);  // clamp; masked at store
        int pi = pc / TILE;
        int pj = pc - pi * TILE;
        xb[k]  = (pi * 16 + pj) * 4;
        acc[k].x = inf; acc[k].y = inf; acc[k].z = inf; acc[k].w = inf;
    }

    #pragma unroll 1
    for (int di = 0; di < 3; ++di) {
        #pragma unroll 1
        for (int dj = 0; dj < 3; ++dj) {
            const int kpos = di * 3 + dj;
            const int xoff = (di * 16 + dj) * 4;             // float4 offset for (di,dj)
            #pragma unroll
            for (int c = 0; c < CIN; c += 4) {
                const int wb = (kpos * CIN + c) * (FOUT / 4) + fg;
                const float4 w0 = WL4[wb];                   // W[.,.,c+0, f..f+3]
                const float4 w1 = WL4[wb + 8];               // c+1
                const float4 w2 = WL4[wb + 16];              // c+2
                const float4 w3 = WL4[wb + 24];              // c+3
                #pragma unroll
                for (int k = 0; k < 7; ++k) {
                    const float4 xv = XS4[xb[k] + xoff + (c >> 2)]; // x[.,pi+di,pj+dj,c..c+3]
                    acc[k].x = min3f(acc[k].x, xv.x - w0.x, xv.y - w1.x);
                    acc[k].x = min3f(acc[k].x, xv.z - w2.x, xv.w - w3.x);
                    acc[k].y = min3f(acc[k].y, xv.x - w0.y, xv.y - w1.y);
                    acc[k].y = min3f(acc[k].y, xv.z - w2.y, xv.w - w3.y);
                    acc[k].z = min3f(acc[k].z, xv.x - w0.z, xv.y - w1.z);
                    acc[k].z = min3f(acc[k].z, xv.z - w2.z, xv.w - w3.z);
                    acc[k].w = min3f(acc[k].w, xv.x - w0.w, xv.y - w1.w);
                    acc[k].w = min3f(acc[k].w, xv.z - w2.w, xv.w - w3.w);
                }
            }
        }
    }

    #pragma unroll
    for (int k = 0; k < 7; ++k) {
        const int p = sp + 32 * k;
        if (p < TILE * TILE) {
            const int pi = p / TILE;
            const int pj = p - pi * TILE;
            const size_t o = ((((size_t)b * HOUT + (i0 + pi)) * WOUT + (j0 + pj)) * FOUT)
                           + (size_t)fg * 4;
            *reinterpret_cast<float4*>(out + o) = acc[k];
        }
    }
}

extern "C" void kernel_launch(void* const* d_in, const int* in_sizes, int n_in,
                              void* d_out, int out_size, void* d_ws, size_t ws_size,
                              hipStream_t stream) {
    (void)in_sizes; (void)n_in; (void)out_size; (void)d_ws; (void)ws_size;
    const float* x  = (const float*)d_in[0];
    const float* w  = (const float*)d_in[1];
    float* out      = (float*)d_out;
    dim3 grid(NTIL, NTIL, BATCH);   // 9 x 9 x 16 = 1296 workgroups
    MorphologicalErosion_62861141344689_kernel<<<grid, dim3(256), 0, stream>>>(x, w, out);
}